// GCN_2551210573869
// MI455X (gfx1250) — compile-verified
//
#include <hip/hip_runtime.h>

typedef __attribute__((ext_vector_type(2))) float v2f;
typedef __attribute__((ext_vector_type(8))) float v8f;

#define N_USER 55485
#define N_ITEM 5986
#define N_NODES (N_USER + N_ITEM)   // 61471
#define NEDGE   1500000
#define F_IN    128
#define SLOPE   0.01f

// ---------------------------------------------------------------------------
// zero-fill scratch (msgs / deg) -- required every call (harness poisons ws)
// ---------------------------------------------------------------------------
__global__ void gcn_zero_kernel(float* __restrict__ p, long n) {
  long i = (long)blockIdx.x * blockDim.x + threadIdx.x;
  if (i < n) p[i] = 0.0f;
}

// ---------------------------------------------------------------------------
// edge scatter: msgs[dst] += x[src] (float4 per lane, 32 lanes = 128 feats),
// optional degree count.  Source rows selected from two tensors (user/item
// concat handled by branch; pass nuser = N_NODES to force single tensor).
// ---------------------------------------------------------------------------
__global__ void gcn_scatter_kernel(const float* __restrict__ xa,
                                   const float* __restrict__ xb,
                                   int nuser,
                                   const int* __restrict__ src,
                                   const int* __restrict__ dst,
                                   float* __restrict__ msgs,
                                   float* __restrict__ deg,
                                   int count_deg) {
  long idx = (long)blockIdx.x * blockDim.x + threadIdx.x;
  long e = idx >> 5;
  if (e >= NEDGE) return;
  int lane = (int)(idx & 31);
  int f = lane << 2;                       // 4 floats per lane
  int s = src[e];
  int d = dst[e];
  const float* xr = (s < nuser) ? (xa + (size_t)s * F_IN)
                                : (xb + (size_t)(s - nuser) * F_IN);
  float4 v = *(const float4*)(xr + f);
  float* m = msgs + (size_t)d * F_IN + f;
  atomicAdd(m + 0, v.x);
  atomicAdd(m + 1, v.y);
  atomicAdd(m + 2, v.z);
  atomicAdd(m + 3, v.w);
  if (count_deg && lane == 0) atomicAdd(deg + d, 1.0f);
}

// ---------------------------------------------------------------------------
// Fused SAGE layer:  out = act( (msgs/deg) @ Wl + bias + x @ Wr )
// One wave per 16x16 output tile, V_WMMA_F32_16X16X4_F32, K = 128.
// blockDim.x = (OUTF/16) waves * 32; gridDim.x = ceil(N/16) row tiles.
// ---------------------------------------------------------------------------
template <int OUTF, bool LEAKY>
__global__ void gcn_sage_wmma_kernel(const float* __restrict__ msgs,
                                     const float* __restrict__ deg,
                                     const float* __restrict__ xa,
                                     const float* __restrict__ xb,
                                     int nuser,
                                     const float* __restrict__ Wl,
                                     const float* __restrict__ Wr,
                                     const float* __restrict__ bias,
                                     float* __restrict__ out) {
  const int lane = threadIdx.x & 31;
  const int wave = threadIdx.x >> 5;
  const int n0 = wave * 16;                 // output-column tile base
  const int row_base = blockIdx.x * 16;     // output-row tile base

  // --- A-fragment addressing (16x4 f32: lane L -> row L&15, k-pair (L>>4)*2)
  const int rlocal = lane & 15;
  int g = row_base + rlocal;
  if (g >= N_NODES) g = N_NODES - 1;        // clamp: keep EXEC all-ones, mask at store
  const int khalf = (lane >> 4) << 1;

  const float* mrow = msgs + (size_t)g * F_IN;
  const float* xrow = (g < nuser) ? (xa + (size_t)g * F_IN)
                                  : (xb + (size_t)(g - nuser) * F_IN);
  const float rdeg = 1.0f / fmaxf(deg[g], 1.0f);

  // --- B-fragment addressing (4x16 f32: lane L -> col L&15, k-pair (L>>4)*2)
  const int ncol = lane & 15;
  const float* wl = Wl + n0 + ncol;
  const float* wr = Wr + n0 + ncol;

  v8f acc = {};
#pragma unroll 8
  for (int k = 0; k < F_IN; k += 4) {
    const int kk = k + khalf;
    v2f a_mean, a_self, b_l, b_r;
    a_mean.x = mrow[kk] * rdeg;
    a_mean.y = mrow[kk + 1] * rdeg;
    a_self.x = xrow[kk];
    a_self.y = xrow[kk + 1];
    b_l.x = wl[(size_t)kk * OUTF];
    b_l.y = wl[(size_t)(kk + 1) * OUTF];
    b_r.x = wr[(size_t)kk * OUTF];
    b_r.y = wr[(size_t)(kk + 1) * OUTF];
    // D = A*B + C  (fp32 WMMA, 8 args: neg_a, A, neg_b, B, c_mod, C, reuseA, reuseB)
    acc = __builtin_amdgcn_wmma_f32_16x16x4_f32(false, a_mean, false, b_l,
                                                (short)0, acc, false, false);
    acc = __builtin_amdgcn_wmma_f32_16x16x4_f32(false, a_self, false, b_r,
                                                (short)0, acc, false, false);
  }

  // --- epilogue: bias + activation + masked store
  // C/D layout: VGPR i, lane L -> row = i + (L>>4)*8, col = L&15
  const float bv = bias[n0 + ncol];
  const int mbase = (lane >> 4) << 3;
#pragma unroll
  for (int i = 0; i < 8; ++i) {
    int grow = row_base + mbase + i;
    if (grow < N_NODES) {
      float v = acc[i] + bv;
      if (LEAKY) v = (v > 0.0f) ? v : v * SLOPE;
      out[(size_t)grow * OUTF + n0 + ncol] = v;
    }
  }
}

// ---------------------------------------------------------------------------
extern "C" void kernel_launch(void* const* d_in, const int* in_sizes, int n_in,
                              void* d_out, int out_size, void* d_ws, size_t ws_size,
                              hipStream_t stream) {
  const float* user = (const float*)d_in[0];
  const float* item = (const float*)d_in[1];
  const float* W1l  = (const float*)d_in[2];
  const float* b1   = (const float*)d_in[3];
  const float* W1r  = (const float*)d_in[4];
  const float* W2l  = (const float*)d_in[5];
  const float* b2   = (const float*)d_in[6];
  const float* W2r  = (const float*)d_in[7];
  const int*   ei   = (const int*)d_in[8];
  const int* src = ei;            // edge_index[0]
  const int* dst = ei + NEDGE;    // edge_index[1]

  // workspace layout (floats): [deg | msgs | h1]
  float* ws = (float*)d_ws;
  const long DEG_SZ  = 61504;                       // N rounded up to 64
  const long MSG_SZ  = (long)N_NODES * F_IN;        // 7,868,288
  float* deg  = ws;
  float* msgs = ws + DEG_SZ;
  float* h1   = ws + DEG_SZ + MSG_SZ;

  const int row_tiles = (N_NODES + 15) / 16;        // 3842
  const long scat_threads = (long)NEDGE * 32;
  const int scat_blocks = (int)((scat_threads + 255) / 256);

  // 1) zero deg + msgs
  {
    long n = DEG_SZ + MSG_SZ;
    gcn_zero_kernel<<<(int)((n + 255) / 256), 256, 0, stream>>>(ws, n);
  }
  // 2) layer-1 scatter (+degree)
  gcn_scatter_kernel<<<scat_blocks, 256, 0, stream>>>(
      user, item, N_USER, src, dst, msgs, deg, 1);
  // 3) layer-1 fused GEMM + bias + leaky -> h1   (128 -> 128, 8 waves/block)
  gcn_sage_wmma_kernel<128, true><<<row_tiles, 256, 0, stream>>>(
      msgs, deg, user, item, N_USER, W1l, W1r, b1, h1);
  // 4) re-zero msgs (degree unchanged)
  gcn_zero_kernel<<<(int)((MSG_SZ + 255) / 256), 256, 0, stream>>>(msgs, MSG_SZ);
  // 5) layer-2 scatter from h1
  gcn_scatter_kernel<<<scat_blocks, 256, 0, stream>>>(
      h1, h1, N_NODES, src, dst, msgs, deg, 0);
  // 6) layer-2 fused GEMM + bias -> out          (128 -> 64, 4 waves/block)
  gcn_sage_wmma_kernel<64, false><<<row_tiles, 128, 0, stream>>>(
      msgs, deg, h1, h1, N_NODES, W2l, W2r, b2, (float*)d_out);
}